// SpatialTransformation_73701638799380
// MI455X (gfx1250) — compile-verified
//
#include <hip/hip_runtime.h>
#include <hip/hip_bf16.h>

// 3D trilinear backward warp (spatial transformer) for MI455X (gfx1250).
//
// Roofline: ~113MB deformation (stream, 0 reuse) + ~37.5MB output (stream)
// + ~75MB image (8x gather reuse, fully L2-resident in the 192MB L2)
// => ~225MB HBM => ~10us at 23.3 TB/s. Pure bandwidth problem; no matmul
// structure, so no WMMA. The levers on this chip are cache policy (TH hints)
// and minimizing per-voxel instruction issue.
//
// This version: 3D launch (no div/mod), branchless zero-pad via per-axis
// weight zeroing (no exec-mask branches), 32-bit saddr+voffset addressing,
// NT hints on streams, global_prefetch_b8 on the deformation stream.

#define B_ 2
#define H_ 160
#define W_ 192
#define D_ 160

__global__ void __launch_bounds__(320)
SpatialTransformation_73701638799380_kernel(const float* __restrict__ img,
                                            const float* __restrict__ def,
                                            float* __restrict__ out) {
    const int d = (int)threadIdx.x;                         // lanes span D
    const int w = (int)(blockIdx.x * 2 + threadIdx.y);
    const int h = (int)blockIdx.y;
    const int b = (int)blockIdx.z;
    const int idx = ((b * H_ + h) * W_ + w) * D_ + d;       // < 2^24, fits i32

    // Prefetch next row's deformation chunk into L2 (stream has zero reuse).
    if (h + 1 < H_) {
        __builtin_prefetch(def + (size_t)(idx + W_ * D_) * 3, 0, 0);
    }

    // Deformation: 3 contiguous floats/voxel, streamed once -> NT load(s).
    const size_t dbase = (size_t)idx * 3;
    const float dxv = __builtin_nontemporal_load(def + dbase + 0);
    const float dyv = __builtin_nontemporal_load(def + dbase + 1);
    const float dzv = __builtin_nontemporal_load(def + dbase + 2);

    // Sampling coords in padded space (+1 zero-pad offset), as in reference.
    const float xf = dxv + (float)w + 1.0f;
    const float yf = dyv + (float)h + 1.0f;
    const float zf = dzv + (float)d + 1.0f;

    const int x0 = (int)floorf(xf);     // padded-space lower corners
    const int y0 = (int)floorf(yf);
    const int z0 = (int)floorf(zf);

    // Weights from unclamped corners. Equivalent to the reference's
    // clamp-then-subtract: clamping only alters weights of corners whose
    // contribution is zeroed below (pad cells are 0 in the reference).
    float wx = (float)(x0 + 1) - xf;  float ux = 1.0f - wx;
    float wy = (float)(y0 + 1) - yf;  float uy = 1.0f - wy;
    float wz = (float)(z0 + 1) - zf;  float uz = 1.0f - wz;

    // Corner-plane validity: padded coord p is a real voxel iff p in [1,dim].
    // Lower corner plane = x0 (interior index x0-1), upper = x0+1 (interior x0).
    wx = ((unsigned)(x0 - 1) < (unsigned)W_) ? wx : 0.0f;
    ux = ((unsigned)(x0)     < (unsigned)W_) ? ux : 0.0f;
    wy = ((unsigned)(y0 - 1) < (unsigned)H_) ? wy : 0.0f;
    uy = ((unsigned)(y0)     < (unsigned)H_) ? uy : 0.0f;
    wz = ((unsigned)(z0 - 1) < (unsigned)D_) ? wz : 0.0f;
    uz = ((unsigned)(z0)     < (unsigned)D_) ? uz : 0.0f;

    // In-range interior indices (value irrelevant when the weight is zeroed).
    const int ox0 = min(max(x0 - 1, 0), W_ - 1);
    const int ox1 = min(max(x0,     0), W_ - 1);
    const int oy0 = min(max(y0 - 1, 0), H_ - 1);
    const int oy1 = min(max(y0,     0), H_ - 1);
    const int oz0 = min(max(z0 - 1, 0), D_ - 1);
    const int oz1 = min(max(z0,     0), D_ - 1);

    // 32-bit element offsets (image has 9.8M elements -> fits easily).
    const int rb = b * H_;
    const int r0 = (rb + oy0) * W_;
    const int r1 = (rb + oy1) * W_;
    const int a00 = (r0 + ox0) * D_;    // (y0, x0)
    const int a01 = (r0 + ox1) * D_;    // (y0, x1)
    const int a10 = (r1 + ox0) * D_;    // (y1, x0)
    const int a11 = (r1 + ox1) * D_;    // (y1, x1)

    // 8 unconditional gathers (image stays hot in L2; regular RT hint).
    const float Ia = img[a00 + oz0];    // (y0,x0,z0)
    const float Ie = img[a00 + oz1];    // (y0,x0,z1)
    const float Ic = img[a01 + oz0];    // (y0,x1,z0)
    const float Ig = img[a01 + oz1];    // (y0,x1,z1)
    const float Ib = img[a10 + oz0];    // (y1,x0,z0)
    const float If = img[a10 + oz1];    // (y1,x0,z1)
    const float Id = img[a11 + oz0];    // (y1,x1,z0)
    const float Ih = img[a11 + oz1];    // (y1,x1,z1)

    // Factored trilinear blend: 14 mul/fma instead of 32.
    const float r =
        wy * (wx * (wz * Ia + uz * Ie) + ux * (wz * Ic + uz * Ig)) +
        uy * (wx * (wz * Ib + uz * If) + ux * (wz * Id + uz * Ih));

    // Output streamed once -> NT store (keep L2 for the image).
    __builtin_nontemporal_store(r, out + idx);
}

extern "C" void kernel_launch(void* const* d_in, const int* in_sizes, int n_in,
                              void* d_out, int out_size, void* d_ws, size_t ws_size,
                              hipStream_t stream) {
    const float* img = (const float*)d_in[0];  // moving_image  [2,160,192,160] f32
    const float* def = (const float*)d_in[1];  // deformation   [2,160,192,160,3] f32
    float* out = (float*)d_out;                // warped        [2,160,192,160] f32

    // D along threads (160 = 5 wave32 waves), 2 W-columns per block -> 320 thr.
    dim3 block(D_, 2, 1);
    dim3 grid(W_ / 2, H_, B_);
    SpatialTransformation_73701638799380_kernel<<<grid, block, 0, stream>>>(img, def, out);
}